// ConcatCriticTorch_86620900426436
// MI455X (gfx1250) — compile-verified
//
#include <hip/hip_runtime.h>
#include <hip/hip_bf16.h>

typedef __attribute__((ext_vector_type(2))) float v2f;
typedef __attribute__((ext_vector_type(8))) float v8f;

constexpr int BN = 512;  // batch
constexpr int DD = 128;  // per-input feature dim
constexpr int HH = 512;  // hidden dim

// ---------------------------------------------------------------------------
// Kernel 1: Hx = X @ W1[0:128,:]         (z == 0)
//           Hy = Y @ W1[128:256,:] + b1  (z == 1)
// One wave (32 lanes) computes one 16x16 f32 tile with V_WMMA_F32_16X16X4_F32,
// looping K=128 in steps of 4 (32 WMMAs per tile).
//
// A-fragment layout (ISA 7.12.2, 32-bit A 16x4): lanes 0-15 hold row M=l,
// K={k0,k0+1}; lanes 16-31 hold row M=l, K={k0+2,k0+3}. Row-major X makes
// this a single contiguous float2 load per lane per K-step.
// B-fragment: same K split across lane halves, N striped across lanes.
// C/D: VGPR r -> row (r + 8*half), N = lane&15.
// ---------------------------------------------------------------------------
__global__ __launch_bounds__(256) void gemm_hidden_wmma(
    const float* __restrict__ X, const float* __restrict__ Y,
    const float* __restrict__ W1, const float* __restrict__ b1,
    float* __restrict__ Hx, float* __restrict__ Hy) {
  const int z = blockIdx.z;
  const float* __restrict__ In = z ? Y : X;
  const float* __restrict__ W  = W1 + (z ? DD * HH : 0);
  float* __restrict__ Out      = z ? Hy : Hx;

  const int lane = threadIdx.x & 31;
  const int wave = threadIdx.x >> 5;
  const int tile = blockIdx.x * 8 + wave;   // 1024 16x16 tiles total
  const int tM = tile >> 5;                 // 32 tiles along M
  const int tN = tile & 31;                 // 32 tiles along N
  const int half = lane >> 4;               // 0: K lo pair, 1: K hi pair
  const int l = lane & 15;

  v8f c = {0.f, 0.f, 0.f, 0.f, 0.f, 0.f, 0.f, 0.f};
  const float* __restrict__ arow = In + (tM * 16 + l) * DD + half * 2;
  const float* __restrict__ bcol = W + (half * 2) * HH + tN * 16 + l;

  #pragma unroll 4
  for (int k = 0; k < DD; k += 4) {
    const float2 a2 = *(const float2*)(arow + k);
    v2f a, b;
    a.x = a2.x;             // A[m, k + 2*half]
    a.y = a2.y;             // A[m, k + 2*half + 1]
    b.x = bcol[k * HH];          // B[k + 2*half,     n]
    b.y = bcol[k * HH + HH];     // B[k + 2*half + 1, n]
    c = __builtin_amdgcn_wmma_f32_16x16x4_f32(
        /*neg_a=*/false, a, /*neg_b=*/false, b,
        /*c_mod=*/(short)0, c, /*reuse_a=*/false, /*reuse_b=*/false);
  }

  const int col = tN * 16 + l;
  const float bias = z ? b1[col] : 0.0f;
  #pragma unroll
  for (int r = 0; r < 8; ++r) {
    const int row = tM * 16 + half * 8 + r;
    Out[row * HH + col] = c[r] + bias;
  }
}

// ---------------------------------------------------------------------------
// Kernel 2: out[a,b] = sum_h relu(Hx[a,h] + Hy[b,h]) * w2[h] + b2
// 256-thread block -> 64x64 output tile, 4x4 register blocking per thread.
// H chunked by 32 through LDS; row stride 33 floats kills bank conflicts
// (stride-4 row access -> bank stride 4 across the 16 tx lanes).
// L2-resident working set (Hx+Hy = 2 MB << 192 MB) -> pure VALU bound.
// ---------------------------------------------------------------------------
__global__ __launch_bounds__(256) void pair_score(
    const float* __restrict__ Hx, const float* __restrict__ Hy,
    const float* __restrict__ W2, const float* __restrict__ b2,
    float* __restrict__ out) {
  __shared__ float hxs[64 * 33];
  __shared__ float hys[64 * 33];
  __shared__ float w2s[32];

  const int t  = threadIdx.x;
  const int tx = t & 15;   // b direction (4 cols each)
  const int ty = t >> 4;   // a direction (4 rows each)
  const int a0 = blockIdx.y * 64;
  const int b0 = blockIdx.x * 64;

  const int lrow = t >> 2;        // 0..63: staged row
  const int jb   = (t & 3) * 8;   // 0,8,16,24: col base (8 floats per thread)

  float acc[4][4] = {};

  for (int h0 = 0; h0 < HH; h0 += 32) {
    // Stage into registers before the barrier so loads overlap prior compute.
    const float4 vx0 = *(const float4*)(Hx + (a0 + lrow) * HH + h0 + jb);
    const float4 vx1 = *(const float4*)(Hx + (a0 + lrow) * HH + h0 + jb + 4);
    const float4 vy0 = *(const float4*)(Hy + (b0 + lrow) * HH + h0 + jb);
    const float4 vy1 = *(const float4*)(Hy + (b0 + lrow) * HH + h0 + jb + 4);
    const float wv = (t < 32) ? W2[h0 + t] : 0.0f;
    __syncthreads();
    float* px = hxs + lrow * 33 + jb;
    px[0] = vx0.x; px[1] = vx0.y; px[2] = vx0.z; px[3] = vx0.w;
    px[4] = vx1.x; px[5] = vx1.y; px[6] = vx1.z; px[7] = vx1.w;
    float* py = hys + lrow * 33 + jb;
    py[0] = vy0.x; py[1] = vy0.y; py[2] = vy0.z; py[3] = vy0.w;
    py[4] = vy1.x; py[5] = vy1.y; py[6] = vy1.z; py[7] = vy1.w;
    if (t < 32) w2s[t] = wv;
    __syncthreads();

    #pragma unroll 4
    for (int h = 0; h < 32; ++h) {
      const float w = w2s[h];
      float xa[4], yb[4];
      #pragma unroll
      for (int r = 0; r < 4; ++r) xa[r] = hxs[(ty * 4 + r) * 33 + h];
      #pragma unroll
      for (int cI = 0; cI < 4; ++cI) yb[cI] = hys[(tx * 4 + cI) * 33 + h];
      #pragma unroll
      for (int r = 0; r < 4; ++r)
        #pragma unroll
        for (int cI = 0; cI < 4; ++cI) {
          float s = xa[r] + yb[cI];
          s = fmaxf(s, 0.0f);
          acc[r][cI] = fmaf(s, w, acc[r][cI]);
        }
    }
  }

  const float bb = b2[0];
  #pragma unroll
  for (int r = 0; r < 4; ++r)
    #pragma unroll
    for (int cI = 0; cI < 4; ++cI)
      out[(a0 + ty * 4 + r) * BN + (b0 + tx * 4 + cI)] = acc[r][cI] + bb;
}

extern "C" void kernel_launch(void* const* d_in, const int* in_sizes, int n_in,
                              void* d_out, int out_size, void* d_ws, size_t ws_size,
                              hipStream_t stream) {
  const float* x  = (const float*)d_in[0];   // [512,128]
  const float* y  = (const float*)d_in[1];   // [512,128]
  const float* W1 = (const float*)d_in[2];   // [256,512]
  const float* b1 = (const float*)d_in[3];   // [512]
  const float* W2 = (const float*)d_in[4];   // [512,1]
  const float* b2 = (const float*)d_in[5];   // [1]
  float* out = (float*)d_out;                // [512,512]

  float* Hx = (float*)d_ws;                  // 512*512 f32 = 1 MB
  float* Hy = Hx + BN * HH;                  // 1 MB

  // 1024 tiles per GEMM, 8 waves (tiles) per 256-thread block, z selects X/Y.
  dim3 g1(128, 1, 2);
  gemm_hidden_wmma<<<g1, 256, 0, stream>>>(x, y, W1, b1, Hx, Hy);

  // 8x8 blocks of 64x64 output tiles.
  dim3 g2(8, 8, 1);
  pair_score<<<g2, 256, 0, stream>>>(Hx, Hy, W2, b2, out);
}